// RCD_43224550867563
// MI455X (gfx1250) — compile-verified
//
#include <hip/hip_runtime.h>
#include <math.h>

// ---------------------------------------------------------------------------
// RCD forward for MI455X (gfx1250, wave32).
// Dense GEMMs -> V_WMMA_F32_16X16X4_F32 (exact f32 tensor path), K-loop
// software-pipelined in chunks of 32 so 16 b64 fragment loads are in flight
// before the WMMA chain consumes them (instead of s_wait_loadcnt 0 per WMMA).
// Graph softmax/aggregation -> L2-resident f32 atomics, one wave per edge.
// ---------------------------------------------------------------------------

typedef __attribute__((ext_vector_type(2))) float v2f;
typedef __attribute__((ext_vector_type(8))) float v8f;

__device__ __forceinline__ float sigmoidf_(float x) { return 1.0f / (1.0f + expf(-x)); }

__device__ __forceinline__ float wave_sum32(float v) {
#pragma unroll
  for (int off = 16; off > 0; off >>= 1) v += __shfl_xor(v, off, 32);
  return v;
}

// float atomic max via signed/unsigned integer ordering trick (init = -inf)
__device__ __forceinline__ void atomicMaxF(float* addr, float v) {
  if (v >= 0.0f) atomicMax((int*)addr, __float_as_int(v));
  else           atomicMin((unsigned int*)addr, __float_as_uint(v));
}

// ---------------------------------------------------------------------------
__global__ __launch_bounds__(256) void k_zero(float* __restrict__ p, size_t n) {
  size_t i = (size_t)blockIdx.x * blockDim.x + threadIdx.x;
  size_t s = (size_t)gridDim.x * blockDim.x;
  for (; i < n; i += s) p[i] = 0.0f;
}

// ---------------------------------------------------------------------------
// out[m, n] = sum_k A[row(m), k] * W[n*ldw + wofs + k]   (i.e. out = A @ Wslice^T)
// row(m): m -> gather[m] (optional) -> A1 if < n1 else A2 (virtual concat).
// Tile: 16 rows per block; each of 8 waves owns 16-column tiles, K-step = 4
// via V_WMMA_F32_16X16X4_F32.
// Fragment mapping (ISA 7.12.2): lo = lane&15, hi = lane>>4.
//   A frag  a[v]  = Arow[m0+lo][k0 + 2*hi + v]
//   B frag  b[v]  = W[(n0+lo)*ldw + wofs + k0 + 2*hi + v]
//   D frag  d[r] -> out[m0 + r + 8*hi][n0 + lo]
// ---------------------------------------------------------------------------
__global__ __launch_bounds__(256) void k_wmma_gemm(
    const float* __restrict__ A1, int n1, const float* __restrict__ A2,
    const int* __restrict__ gather, int N, int K,
    const float* __restrict__ W, int ldw, int wofs,
    float* __restrict__ out)
{
  const int wave = threadIdx.x >> 5;
  const int lane = threadIdx.x & 31;
  const int lo = lane & 15, hi = lane >> 4;
  const int m0 = blockIdx.x * 16;

  // per-lane source row (clamped; stores are guarded)
  int r = m0 + lo; if (r >= N) r = N - 1;
  int ar = gather ? gather[r] : r;
  const float* arow = (ar < n1) ? (A1 + (size_t)ar * K)
                                : (A2 + (size_t)(ar - n1) * K);

  for (int n0 = wave * 16; n0 < K; n0 += 128) {   // 8 waves * 16 cols
    const float* wrow = W + (size_t)(n0 + lo) * ldw + wofs;
    v8f acc = {};
    int k0 = 0;
    // pipelined main loop: issue 16 b64 loads, then 8 dependent WMMAs
    for (; k0 + 32 <= K; k0 += 32) {
      v2f a[8], b[8];
#pragma unroll
      for (int u = 0; u < 8; ++u) {
        const int kk = k0 + 4 * u + 2 * hi;
        a[u].x = arow[kk]; a[u].y = arow[kk + 1];
        b[u].x = wrow[kk]; b[u].y = wrow[kk + 1];
      }
#pragma unroll
      for (int u = 0; u < 8; ++u)
        acc = __builtin_amdgcn_wmma_f32_16x16x4_f32(
            false, a[u], false, b[u], (short)0, acc, false, false);
    }
    // remainder (K % 32)
    for (; k0 < K; k0 += 4) {
      const int kk = k0 + 2 * hi;
      v2f a; a.x = arow[kk]; a.y = arow[kk + 1];
      v2f b; b.x = wrow[kk]; b.y = wrow[kk + 1];
      acc = __builtin_amdgcn_wmma_f32_16x16x4_f32(
          false, a, false, b, (short)0, acc, false, false);
    }
#pragma unroll
    for (int rr = 0; rr < 8; ++rr) {
      const int m = m0 + rr + 8 * hi;
      if (m < N) out[(size_t)m * K + n0 + lo] = acc[rr];
    }
  }
}

// ---------------------------------------------------------------------------
// Per-node: es = z . a[0:K], ed = z . a[K:2K]; init m=-inf, denom=0.
// One wave per node.
// ---------------------------------------------------------------------------
__global__ __launch_bounds__(256) void k_node_dots(
    const float* __restrict__ z, const float* __restrict__ a, int N, int K,
    float* __restrict__ es, float* __restrict__ ed,
    float* __restrict__ m, float* __restrict__ denom)
{
  const int n = (int)((blockIdx.x * blockDim.x + threadIdx.x) >> 5);
  const int lane = threadIdx.x & 31;
  if (n >= N) return;
  const float* zr = z + (size_t)n * K;
  float s1 = 0.0f, s2 = 0.0f;
  for (int c = lane; c < K; c += 32) {
    const float zv = zr[c];
    s1 += zv * a[c];
    s2 += zv * a[K + c];
  }
  s1 = wave_sum32(s1);
  s2 = wave_sum32(s2);
  if (lane == 0) {
    es[n] = s1; ed[n] = s2;
    m[n] = -__builtin_huge_valf();
    denom[n] = 0.0f;
  }
}

__global__ __launch_bounds__(256) void k_edge_max(
    const int* __restrict__ src, const int* __restrict__ dst, int E,
    const float* __restrict__ es, const float* __restrict__ ed,
    float* __restrict__ m)
{
  for (int e = blockIdx.x * blockDim.x + threadIdx.x; e < E;
       e += gridDim.x * blockDim.x) {
    const int d = dst[e];
    atomicMaxF(&m[d], es[src[e]] + ed[d]);
  }
}

__global__ __launch_bounds__(256) void k_edge_den(
    const int* __restrict__ src, const int* __restrict__ dst, int E,
    const float* __restrict__ es, const float* __restrict__ ed,
    const float* __restrict__ m, float* __restrict__ denom)
{
  for (int e = blockIdx.x * blockDim.x + threadIdx.x; e < E;
       e += gridDim.x * blockDim.x) {
    const int d = dst[e];
    atomicAdd(&denom[d], expf(es[src[e]] + ed[d] - m[d]));
  }
}

// One wave per edge: out[dst-dstLo, :] += alpha * z[src, :]
__global__ __launch_bounds__(256) void k_edge_agg(
    const int* __restrict__ src, const int* __restrict__ dst, int E,
    const float* __restrict__ es, const float* __restrict__ ed,
    const float* __restrict__ m, const float* __restrict__ denom,
    const float* __restrict__ z, int K, float* __restrict__ out, int dstLo)
{
  const int gw = (int)((blockIdx.x * blockDim.x + threadIdx.x) >> 5);
  const int lane = threadIdx.x & 31;
  const int nw = (int)((gridDim.x * blockDim.x) >> 5);
  for (int e = gw; e < E; e += nw) {
    const int s = src[e], d = dst[e];
    const float alpha = expf(es[s] + ed[d] - m[d]) / denom[d];
    const float* zr = z + (size_t)s * K;
    float* o = out + (size_t)(d - dstLo) * K;
    for (int c = lane; c < K; c += 32) atomicAdd(&o[c], alpha * zr[c]);
  }
}

// ---------------------------------------------------------------------------
// kn_new = kn + softmax3(s1,s2,s3) . (B, C, D), one wave per row.
// s_i = kn.row · w_i[:K] + X.row · w_i[K:] + b_i
// ---------------------------------------------------------------------------
__global__ __launch_bounds__(256) void k_update_kn(
    const float* __restrict__ kn,
    const float* __restrict__ Bm, const float* __restrict__ Cm,
    const float* __restrict__ Dm,
    const float* __restrict__ w1, const float* __restrict__ b1,
    const float* __restrict__ w2, const float* __restrict__ b2,
    const float* __restrict__ w3, const float* __restrict__ b3,
    int K, float* __restrict__ out)
{
  const int n = (int)((blockIdx.x * blockDim.x + threadIdx.x) >> 5);
  const int lane = threadIdx.x & 31;
  if (n >= K) return;
  const float* kr = kn + (size_t)n * K;
  const float* br = Bm + (size_t)n * K;
  const float* cr = Cm + (size_t)n * K;
  const float* dr = Dm + (size_t)n * K;
  float p1 = 0, p2 = 0, p3 = 0;
  for (int c = lane; c < K; c += 32) {
    const float kv = kr[c];
    p1 += kv * w1[c] + br[c] * w1[K + c];
    p2 += kv * w2[c] + cr[c] * w2[K + c];
    p3 += kv * w3[c] + dr[c] * w3[K + c];
  }
  float s1 = wave_sum32(p1) + b1[0];
  float s2 = wave_sum32(p2) + b2[0];
  float s3 = wave_sum32(p3) + b3[0];
  const float mx = fmaxf(s1, fmaxf(s2, s3));
  const float e1 = expf(s1 - mx), e2 = expf(s2 - mx), e3 = expf(s3 - mx);
  const float inv = 1.0f / (e1 + e2 + e3);
  const float c1 = e1 * inv, c2 = e2 * inv, c3 = e3 * inv;
  for (int c = lane; c < K; c += 32)
    out[(size_t)n * K + c] = kr[c] + c1 * br[c] + c2 * cr[c] + c3 * dr[c];
}

__global__ __launch_bounds__(256) void k_update_ex(
    const float* __restrict__ ex,
    const float* __restrict__ Be, const float* __restrict__ Ce,
    const float* __restrict__ w1, const float* __restrict__ b1,
    const float* __restrict__ w2, const float* __restrict__ b2,
    int K, int N, float* __restrict__ out)
{
  const int n = (int)((blockIdx.x * blockDim.x + threadIdx.x) >> 5);
  const int lane = threadIdx.x & 31;
  if (n >= N) return;
  const float* xr = ex + (size_t)n * K;
  const float* br = Be + (size_t)n * K;
  const float* cr = Ce + (size_t)n * K;
  float p1 = 0, p2 = 0;
  for (int c = lane; c < K; c += 32) {
    const float xv = xr[c];
    p1 += xv * w1[c] + br[c] * w1[K + c];
    p2 += xv * w2[c] + cr[c] * w2[K + c];
  }
  float t1 = wave_sum32(p1) + b1[0];
  float t2 = wave_sum32(p2) + b2[0];
  const float mx = fmaxf(t1, t2);
  const float e1 = expf(t1 - mx), e2 = expf(t2 - mx);
  const float inv = 1.0f / (e1 + e2);
  const float c1 = e1 * inv, c2 = e2 * inv;
  for (int c = lane; c < K; c += 32)
    out[(size_t)n * K + c] = xr[c] + c1 * br[c] + c2 * cr[c];
}

__global__ __launch_bounds__(256) void k_add(
    const float* __restrict__ a, const float* __restrict__ b, size_t n,
    float* __restrict__ out)
{
  size_t i = (size_t)blockIdx.x * blockDim.x + threadIdx.x;
  size_t s = (size_t)gridDim.x * blockDim.x;
  for (; i < n; i += s) out[i] = a[i] + b[i];
}

// ---------------------------------------------------------------------------
// out[b] = sum_i sigmoid( sum_c (sig(su[b,c]+knW1[i,c]) - sig(eu[b,c]+knW2[i,c]))
//                         * p3w[c] + p3b ) * kn_r[b,i]  /  sum_i kn_r[b,i]
// One block (8 waves) per b; wave handles rows i = wave, wave+8, ...
// ---------------------------------------------------------------------------
__global__ __launch_bounds__(256) void k_final(
    const float* __restrict__ su, const float* __restrict__ eu,
    const float* __restrict__ knW1, const float* __restrict__ knW2,
    const float* __restrict__ p3w, const float* __restrict__ p3b,
    const float* __restrict__ knr, int K, float* __restrict__ out)
{
  const int b = blockIdx.x;
  const int wave = threadIdx.x >> 5;
  const int lane = threadIdx.x & 31;
  __shared__ float sSum[8], sCnt[8];
  const float* sub = su + (size_t)b * K;
  const float* eub = eu + (size_t)b * K;
  const float* krb = knr + (size_t)b * K;
  const float bias = p3b[0];
  float wsum = 0.0f, wcnt = 0.0f;
  for (int i = wave; i < K; i += 8) {
    const float* w1r = knW1 + (size_t)i * K;
    const float* w2r = knW2 + (size_t)i * K;
    float p = 0.0f;
    for (int c = lane; c < K; c += 32) {
      const float pref = sigmoidf_(sub[c] + w1r[c]);
      const float diff = sigmoidf_(eub[c] + w2r[c]);
      p += (pref - diff) * p3w[c];
    }
    p = wave_sum32(p);                  // all lanes hold the total
    if (lane == 0) {
      const float rv = krb[i];
      wsum += sigmoidf_(p + bias) * rv;
      wcnt += rv;
    }
  }
  if (lane == 0) { sSum[wave] = wsum; sCnt[wave] = wcnt; }
  __syncthreads();
  if (threadIdx.x == 0) {
    float S = 0, C = 0;
    for (int w = 0; w < 8; ++w) { S += sSum[w]; C += sCnt[w]; }
    out[b] = S / C;
  }
}

// ---------------------------------------------------------------------------
extern "C" void kernel_launch(void* const* d_in, const int* in_sizes, int n_in,
                              void* d_out, int out_size, void* d_ws, size_t ws_size,
                              hipStream_t stream)
{
  (void)n_in; (void)out_size; (void)ws_size;

  // Input indices: JAX pytree leaf order = dict keys sorted at every level.
  enum {
    IN_EXER_ID = 0, IN_EXER_N = 1,
    IN_DIR_DST = 2, IN_DIR_SRC = 3, IN_EFK_DST = 4, IN_EFK_SRC = 5,
    IN_EFU_DST = 6, IN_EFU_SRC = 7, IN_KFE_DST = 8, IN_KFE_SRC = 9,
    IN_UFE_DST = 10, IN_UFE_SRC = 11, IN_UND_DST = 12, IN_UND_SRC = 13,
    IN_KN_R = 14, IN_EX_TABLE = 15,
    IN_F1 = 16, IN_F2 = 38,
    IN_KN_TABLE = 60, IN_P1 = 61, IN_P2 = 62, IN_P3_B = 63, IN_P3_W = 64,
    IN_STU_TABLE = 65, IN_STU_ID = 66
  };
  // fusion-params leaves, sorted: dir.{W,a}, e1_b, e1_w, e2_b, e2_w,
  // efk.{W,a}, efu.{W,a}, k1_b, k1_w, k2_b, k2_w, k3_b, k3_w,
  // kfe.{W,a}, ufe.{W,a}, und.{W,a}
  enum { F_DIR_W = 0, F_DIR_A, F_E1_B, F_E1_W, F_E2_B, F_E2_W,
         F_EFK_W, F_EFK_A, F_EFU_W, F_EFU_A,
         F_K1_B, F_K1_W, F_K2_B, F_K2_W, F_K3_B, F_K3_W,
         F_KFE_W, F_KFE_A, F_UFE_W, F_UFE_A, F_UND_W, F_UND_A };

  auto F = [&](int i) { return (const float*)d_in[i]; };
  auto I = [&](int i) { return (const int*)d_in[i]; };

  const int Bq   = in_sizes[IN_EXER_ID];
  const int K    = in_sizes[IN_KN_R] / Bq;
  const int EXER = in_sizes[IN_EX_TABLE] / K;
  const int STU  = in_sizes[IN_STU_TABLE] / K;
  const int NMAX = EXER + STU;

  const int E_dir = in_sizes[IN_DIR_SRC], E_und = in_sizes[IN_UND_SRC];
  const int E_kfe = in_sizes[IN_KFE_SRC], E_efk = in_sizes[IN_EFK_SRC];
  const int E_ufe = in_sizes[IN_UFE_SRC], E_efu = in_sizes[IN_EFU_SRC];

  // ---- workspace bump allocator (float units, 256B-aligned chunks) ----
  float* ws = (float*)d_ws;
  size_t off = 0;
  auto alloc = [&](size_t n) { float* p = ws + off; off += (n + 63) & ~(size_t)63; return p; };

  float* kn0 = alloc((size_t)K * K);
  float* ex0 = alloc((size_t)EXER * K);
  float* st0 = alloc((size_t)STU * K);
  float* kn1 = alloc((size_t)K * K);
  float* ex1 = alloc((size_t)EXER * K);
  float* st1 = alloc((size_t)STU * K);
  float* z    = alloc((size_t)NMAX * K);
  float* es   = alloc((size_t)NMAX);
  float* ed   = alloc((size_t)NMAX);
  float* mbuf = alloc((size_t)NMAX);
  float* dbuf = alloc((size_t)NMAX);
  float* Bb  = alloc((size_t)K * K);
  float* Cb  = alloc((size_t)K * K);
  float* Db  = alloc((size_t)K * K);
  float* BeB = alloc((size_t)EXER * K);
  float* CeB = alloc((size_t)EXER * K);
  float* UeB = alloc((size_t)STU * K);
  float* suB  = alloc((size_t)Bq * K);
  float* euB  = alloc((size_t)Bq * K);
  float* knW1 = alloc((size_t)K * K);
  float* knW2 = alloc((size_t)K * K);

  const dim3 blk(256);
  auto gmin = [](long n, long cap) { return (unsigned)((n < cap) ? n : cap); };

  // ---- one full GAT layer ----
  auto run_gat = [&](const float* A1, int n1, const float* A2, int N,
                     const float* W, const float* avec,
                     const int* esrc, const int* edst, int E,
                     float* outBuf, int dstLo, long outRows) {
    const size_t on = (size_t)outRows * K;
    k_zero<<<dim3(gmin((long)((on + 255) / 256), 4096)), blk, 0, stream>>>(outBuf, on);
    k_wmma_gemm<<<dim3((N + 15) / 16), blk, 0, stream>>>(
        A1, n1, A2, nullptr, N, K, W, K, 0, z);
    k_node_dots<<<dim3((N + 7) / 8), blk, 0, stream>>>(z, avec, N, K, es, ed, mbuf, dbuf);
    const unsigned ge = gmin((E + 255L) / 256L, 1 << 18);
    k_edge_max<<<dim3(ge), blk, 0, stream>>>(esrc, edst, E, es, ed, mbuf);
    k_edge_den<<<dim3(ge), blk, 0, stream>>>(esrc, edst, E, es, ed, mbuf, dbuf);
    k_edge_agg<<<dim3((E + 7) / 8), blk, 0, stream>>>(
        esrc, edst, E, es, ed, mbuf, dbuf, z, K, outBuf, dstLo);
  };

  // ---- one fusion layer ----
  auto run_fusion = [&](int fb, const float* kn, const float* ex, const float* st,
                        float* nkn, float* nex, float* nst) {
    // dir / und GATs over kn (K nodes)
    run_gat(kn, K, nullptr, K, F(fb + F_DIR_W), F(fb + F_DIR_A),
            I(IN_DIR_SRC), I(IN_DIR_DST), E_dir, Bb, 0, K);
    run_gat(kn, K, nullptr, K, F(fb + F_UND_W), F(fb + F_UND_A),
            I(IN_UND_SRC), I(IN_UND_DST), E_und, Cb, 0, K);
    // kfe / efk over ek = [ex ; kn]  (EXER+K nodes)
    run_gat(ex, EXER, kn, EXER + K, F(fb + F_KFE_W), F(fb + F_KFE_A),
            I(IN_KFE_SRC), I(IN_KFE_DST), E_kfe, Db, EXER, K);      // D = kfe[EXER:]
    run_gat(ex, EXER, kn, EXER + K, F(fb + F_EFK_W), F(fb + F_EFK_A),
            I(IN_EFK_SRC), I(IN_EFK_DST), E_efk, BeB, 0, EXER);     // Be = efk[:EXER]
    // ufe / efu over eu = [ex ; st]  (EXER+STU nodes)
    run_gat(ex, EXER, st, EXER + STU, F(fb + F_UFE_W), F(fb + F_UFE_A),
            I(IN_UFE_SRC), I(IN_UFE_DST), E_ufe, UeB, EXER, STU);   // ufe[EXER:]
    run_gat(ex, EXER, st, EXER + STU, F(fb + F_EFU_W), F(fb + F_EFU_A),
            I(IN_EFU_SRC), I(IN_EFU_DST), E_efu, CeB, 0, EXER);     // Ce = efu[:EXER]
    // table updates
    k_update_kn<<<dim3((K + 7) / 8), blk, 0, stream>>>(
        kn, Bb, Cb, Db,
        F(fb + F_K1_W), F(fb + F_K1_B), F(fb + F_K2_W), F(fb + F_K2_B),
        F(fb + F_K3_W), F(fb + F_K3_B), K, nkn);
    k_update_ex<<<dim3((EXER + 7) / 8), blk, 0, stream>>>(
        ex, BeB, CeB,
        F(fb + F_E1_W), F(fb + F_E1_B), F(fb + F_E2_W), F(fb + F_E2_B),
        K, EXER, nex);
    const size_t sn = (size_t)STU * K;
    k_add<<<dim3(gmin((long)((sn + 255) / 256), 4096)), blk, 0, stream>>>(st, UeB, sn, nst);
  };

  // fusion 1: params -> set0 ; fusion 2: set0 -> set1
  run_fusion(IN_F1, F(IN_KN_TABLE), F(IN_EX_TABLE), F(IN_STU_TABLE), kn0, ex0, st0);
  run_fusion(IN_F2, kn0, ex0, st0, kn1, ex1, st1);

  // ---- prediction head ----
  const float* p1 = F(IN_P1);
  const float* p2 = F(IN_P2);
  // su = st[stu_id] @ W1[:, :K]^T ; knW1 = kn @ W1[:, K:]^T
  k_wmma_gemm<<<dim3((Bq + 15) / 16), blk, 0, stream>>>(
      st1, STU, nullptr, I(IN_STU_ID), Bq, K, p1, 2 * K, 0, suB);
  k_wmma_gemm<<<dim3((K + 15) / 16), blk, 0, stream>>>(
      kn1, K, nullptr, nullptr, K, K, p1, 2 * K, K, knW1);
  // eu = ex[exer_id] @ W2[:, :K]^T ; knW2 = kn @ W2[:, K:]^T
  k_wmma_gemm<<<dim3((Bq + 15) / 16), blk, 0, stream>>>(
      ex1, EXER, nullptr, I(IN_EXER_ID), Bq, K, p2, 2 * K, 0, euB);
  k_wmma_gemm<<<dim3((K + 15) / 16), blk, 0, stream>>>(
      kn1, K, nullptr, nullptr, K, K, p2, 2 * K, K, knW2);

  k_final<<<dim3(Bq), blk, 0, stream>>>(
      suB, euB, knW1, knW2, F(IN_P3_W), F(IN_P3_B), F(IN_KN_R), K, (float*)d_out);
}